// RoPE_85409719649080
// MI455X (gfx1250) — compile-verified
//
#include <hip/hip_runtime.h>

// RoPE on MI455X (gfx1250): bandwidth-bound elementwise rotation.
// Exploits the block-diagonal 2x2 structure of rotation_matrix:
//   y[2k]   = c*x[2k] - s*x[2k+1],  y[2k+1] = s*x[2k] + c*x[2k+1]
// with c = R[pos,2k,2k], s = R[pos,2k+1,2k].
// CDNA5 paths used: GLOBAL_LOAD_ASYNC_TO_LDS_B32 gather (ASYNCcnt),
// s_wait_asynccnt, ds_load_b128 broadcast, non-temporal B128 streaming.

typedef __attribute__((ext_vector_type(4))) float v4f;

#define SEQ_LEN 4096
#define D_K     128
#define N_BH    64            // BATCH*HEADS = 4*16
#define ROW_STRIDE ((size_t)SEQ_LEN * D_K)   // floats between consecutive (b,h) rows

__global__ __launch_bounds__(256) void rope_gfx1250_kernel(
    const float* __restrict__ X,          // (4,16,4096,128) fp32
    const int*   __restrict__ pos_ids,    // (4096,) int32
    const float* __restrict__ R,          // (4096,128,128) fp32, block-diag 2x2
    float*       __restrict__ out)        // (4,16,4096,128) fp32
{
    // Interleaved {c_k, s_k} pairs for k = 0..63  -> 128 floats = 512 B LDS.
    __shared__ __align__(16) float cs[D_K];

    const int s    = blockIdx.x;      // sequence position handled by this block
    const int tid  = threadIdx.x;
    const int lane = tid & 31;
    const int wave = tid >> 5;        // 8 waves per block

    // ---- Wave 0: async-gather the 64 (c,s) pairs from R[pos] into LDS ----
    // c_k lives at flat index pos*16384 + 2k*129, s_k at +128 (row 2k+1, col 2k).
    if (tid < 32) {                   // uniform per wave -> EXEC all-ones inside
        const int    pos      = pos_ids[s];
        const float* rbase    = R + (size_t)pos * (D_K * D_K);
        const unsigned lds_lo = (unsigned)(uintptr_t)cs;   // LDS byte offset
#pragma unroll
        for (int i = 0; i < 2; ++i) {
            const int    k  = lane + i * 32;
            const float* pc = rbase + (size_t)k * 258;     // 2k*129
            const float* ps = pc + D_K;                    // +128
            unsigned oc = lds_lo + (unsigned)k * 8u;       // cs[2k]
            unsigned os = oc + 4u;                         // cs[2k+1]
            asm volatile("global_load_async_to_lds_b32 %0, %1, off"
                         :: "v"(oc), "v"(pc) : "memory");
            asm volatile("global_load_async_to_lds_b32 %0, %1, off"
                         :: "v"(os), "v"(ps) : "memory");
        }
        asm volatile("s_wait_asynccnt 0" ::: "memory");
    }
    __syncthreads();                  // make async-written LDS visible to all waves

    // Lane covers dims 4*lane .. 4*lane+3 => pairs k = 2*lane, 2*lane+1.
    // One ds_load_b128 fetches {c0, s0, c1, s1}.
    const v4f csk = ((const v4f*)cs)[lane];
    const float c0 = csk.x, s0 = csk.y, c1 = csk.z, s1 = csk.w;

    // ---- Stream 8 (b,h) rows per wave: B128 NT load -> rotate -> B128 NT store
    const size_t col = (size_t)s * D_K + (size_t)(lane * 4);
    const float* xp  = X   + (size_t)(wave * 8) * ROW_STRIDE + col;
    float*       op  = out + (size_t)(wave * 8) * ROW_STRIDE + col;

#pragma unroll
    for (int r = 0; r < 8; ++r) {
        const v4f x = __builtin_nontemporal_load(
            (const v4f*)(xp + (size_t)r * ROW_STRIDE));
        v4f y;
        y.x = c0 * x.x - s0 * x.y;
        y.y = s0 * x.x + c0 * x.y;
        y.z = c1 * x.z - s1 * x.w;
        y.w = s1 * x.z + c1 * x.w;
        __builtin_nontemporal_store(y, (v4f*)(op + (size_t)r * ROW_STRIDE));
    }
}

extern "C" void kernel_launch(void* const* d_in, const int* in_sizes, int n_in,
                              void* d_out, int out_size, void* d_ws, size_t ws_size,
                              hipStream_t stream) {
    const float* X   = (const float*)d_in[0];   // (4,16,4096,128) fp32
    const int*   pos = (const int*)d_in[1];     // (4096,) int32
    const float* R   = (const float*)d_in[2];   // (4096,128,128) fp32
    float*       out = (float*)d_out;           // (4,16,4096,128) fp32

    dim3 grid(SEQ_LEN);   // one block per sequence position
    dim3 block(256);      // 8 wave32s
    hipLaunchKernelGGL(rope_gfx1250_kernel, grid, block, 0, stream,
                       X, pos, R, out);
}